// GraphAttentionLayer_635655160695
// MI455X (gfx1250) — compile-verified
//
#include <hip/hip_runtime.h>
#include <stdint.h>

// GAT layer, MI455X (gfx1250), wave32. Matmuls via V_WMMA_F32_16X16X4_F32
// (reference is f32 -> native f32 WMMA path, full precision).
//
// Shapes: B=4, N=2048, F=256, H=8, Dh=32.
// Workspace layout (floats): Wt(64K) | Wh(2M) | e_l(64K) | e_r(64K, mask-
// biased) | m(64K) | sinv(64K) | hp(2M)  ~ 18.1 MB total.

typedef float v2f __attribute__((ext_vector_type(2)));
typedef float v8f __attribute__((ext_vector_type(8)));

#define B_ 4
#define N_ 2048
#define F_ 256
#define H_ 8
#define DH_ 32
#define ALPHA_ 0.2f
#define MASKBIAS -2.0e9f
#define LN_EPS_ 1e-5f

__device__ __forceinline__ float lane_max(float v) {
  #pragma unroll
  for (int m = 16; m >= 1; m >>= 1) v = fmaxf(v, __shfl_xor(v, m, 32));
  return v;
}
__device__ __forceinline__ float lane_sum(float v) {
  #pragma unroll
  for (int m = 16; m >= 1; m >>= 1) v += __shfl_xor(v, m, 32);
  return v;
}

// ---------------- K0: W -> W^T so B-fragments are contiguous b64 loads ------
__global__ void __launch_bounds__(256)
k0_transpose(const float* __restrict__ W, float* __restrict__ Wt) {
  int idx = blockIdx.x * 256 + threadIdx.x;   // 65536 threads
  int r = idx >> 8, c = idx & 255;
  Wt[c * F_ + r] = W[r * F_ + c];
}

// ---------------- K1: Wh = h @ W  (f32 WMMA 16x16x4) ------------------------
// block = 256 thr = 8 waves; wave w -> rows [blk.x*128 + w*16, +16),
// cols [blk.y*32, +32) as two 16x16 accumulators.
__global__ void __launch_bounds__(256)
k1_gemm_wh(const float* __restrict__ h, const float* __restrict__ Wt,
           float* __restrict__ Wh) {
  const int lane = threadIdx.x & 31;
  const int wave = threadIdx.x >> 5;
  const int rowBase = blockIdx.x * 128 + wave * 16;
  const int colBase = blockIdx.y * 32;
  const int lidx = lane & 15;
  const int koff = (lane >> 4) << 1;          // lanes 16-31 carry K+2,K+3

  const float* Aptr = h  + (size_t)(rowBase + lidx) * F_ + koff;
  const float* B0   = Wt + (size_t)(colBase + lidx) * F_ + koff;
  const float* B1   = Wt + (size_t)(colBase + 16 + lidx) * F_ + koff;

  v8f c0 = {}; v8f c1 = {};
  #pragma unroll 4
  for (int k0 = 0; k0 < F_; k0 += 4) {
    v2f a  = *(const v2f*)(Aptr + k0);        // A[M=lidx][k0..k0+1] (+koff)
    v2f b0 = *(const v2f*)(B0 + k0);          // B[k0..k0+1][N=lidx] (+koff)
    v2f b1 = *(const v2f*)(B1 + k0);
    c0 = __builtin_amdgcn_wmma_f32_16x16x4_f32(false, a, false, b0, (short)0, c0, false, false);
    c1 = __builtin_amdgcn_wmma_f32_16x16x4_f32(false, a, false, b1, (short)0, c1, false, false);
  }
  const int rowOut = rowBase + ((lane >> 4) << 3);  // C: VGPR v = row v (+8 hi half)
  #pragma unroll
  for (int v = 0; v < 8; ++v) {
    Wh[(size_t)(rowOut + v) * F_ + colBase + lidx]      = c0[v];
    Wh[(size_t)(rowOut + v) * F_ + colBase + 16 + lidx] = c1[v];
  }
}

// ---------------- K2: e_l/e_r = per-head dot(Wh_row, a); mask folded --------
// Masked columns get e_r += -2e9: after leakyrelu this is ~-4e8 and
// exp(x - m) underflows to exactly 0 == reference's where(mask, -1e9, .).
__global__ void __launch_bounds__(256)
k2_el_er(const float* __restrict__ Wh, const float* __restrict__ a,
         const uint8_t* __restrict__ mask,
         float* __restrict__ e_l, float* __restrict__ e_r) {
  int gw = (blockIdx.x * 256 + threadIdx.x) >> 5;  // wave id: 0..B*N*H-1
  int lane = threadIdx.x & 31;
  int hh = gw & (H_ - 1);
  int bn = gw >> 3;                                 // b*N + n
  float w  = Wh[(size_t)bn * F_ + hh * DH_ + lane];
  float al = a[hh * (2 * DH_) + lane];
  float ar = a[hh * (2 * DH_) + DH_ + lane];
  float sl = lane_sum(w * al);
  float sr = lane_sum(w * ar);
  if (lane == 0) {
    int b = bn / N_, n = bn % N_;
    int o = (b * H_ + hh) * N_ + n;
    e_l[o] = sl;
    e_r[o] = sr + (mask[bn] ? MASKBIAS : 0.0f);
  }
}

// ---------------- K3: softmax stats (row max, 1/sum) ------------------------
// block = 8 waves = 8 rows of one (b,h); biased e_r staged in LDS.
__global__ void __launch_bounds__(256)
k3_stats(const float* __restrict__ e_l, const float* __restrict__ e_r,
         float* __restrict__ m_out, float* __restrict__ sinv_out) {
  __shared__ float ler[N_];
  const int bh = blockIdx.x >> 8;                 // 256 blocks per (b,h)
  const int i0 = (blockIdx.x & 255) * 8;
  for (int j = threadIdx.x; j < N_; j += 256)
    ler[j] = e_r[(size_t)bh * N_ + j];
  __syncthreads();
  const int lane = threadIdx.x & 31;
  const int i = i0 + (threadIdx.x >> 5);
  const float el = e_l[(size_t)bh * N_ + i];
  float m = -3.4e38f;
  for (int j = lane; j < N_; j += 32) {
    float x = el + ler[j];
    m = fmaxf(m, fmaxf(x, ALPHA_ * x));           // leakyrelu = max(x, 0.2x)
  }
  m = lane_max(m);
  float s = 0.f;
  for (int j = lane; j < N_; j += 32) {
    float x = el + ler[j];
    s += __expf(fmaxf(x, ALPHA_ * x) - m);
  }
  s = lane_sum(s);
  if (lane == 0) {
    m_out[(size_t)bh * N_ + i] = m;
    sinv_out[(size_t)bh * N_ + i] = 1.0f / s;
  }
}

// ---------------- K4: h' = softmax(e) @ Wh via f32 WMMA ---------------------
// block = 4 waves, 128-row tile of one (b,h). Each wave owns TWO 16-row
// subtiles (rows r and r+16) and BOTH Dh sub-tiles -> 4 accumulators.
// The two row-sets share the same e_r pair and the same B-fragments, so per
// 4 WMMAs we issue only 3 ds_load_b64 + 4 exps. Scores are built in
// registers directly in WMMA A-fragment layout; Wh tiles staged transposed
// in LDS (stride 66: even -> aligned b64, bank-stride 2 -> conflict-free).
// 1/sum applied to the accumulators once at the end (per-VGPR row scaling).
__global__ void __launch_bounds__(128)
k4_attn_pv(const float* __restrict__ Wh, const float* __restrict__ e_l,
           const float* __restrict__ e_r, const float* __restrict__ m_in,
           const float* __restrict__ sinv, float* __restrict__ hp) {
  __shared__ float ler[N_];
  __shared__ float whT[DH_ * 66];                 // [32 cols][64 j + pad]
  const int bh = blockIdx.x >> 4;                 // 16 row-tiles per (b,h)
  const int i0 = (blockIdx.x & 15) * 128;
  const int b  = bh >> 3;
  const int hh = bh & 7;
  for (int j = threadIdx.x; j < N_; j += 128)
    ler[j] = e_r[(size_t)bh * N_ + j];
  const int lane = threadIdx.x & 31;
  const int wave = threadIdx.x >> 5;              // 4 waves, 32 rows each
  const int lidx = lane & 15;
  const int krow = lane >> 4;                     // hi half carries K+2,K+3
  const int irowA = i0 + wave * 32 + lidx;        // row-set A: r .. r+15
  const int irowB = irowA + 16;                   // row-set B: r+16 .. r+31
  const float elA = e_l[(size_t)bh * N_ + irowA];
  const float miA = m_in[(size_t)bh * N_ + irowA];
  const float elB = e_l[(size_t)bh * N_ + irowB];
  const float miB = m_in[(size_t)bh * N_ + irowB];
  const int sd = threadIdx.x & 31;                // staging: col 0..31
  const int sj = threadIdx.x >> 5;                // staging: j start 0..3
  const float* whBase = Wh + (size_t)b * N_ * F_ + hh * DH_ + sd;
  v8f aA0 = {}; v8f aA1 = {}; v8f aB0 = {}; v8f aB1 = {};
  for (int jb = 0; jb < N_; jb += 64) {
    if (jb + 64 < N_)                             // prefetch next chunk
      __builtin_prefetch(whBase + (size_t)(jb + 64 + sj) * F_, 0, 1);
    __syncthreads();                              // protect previous tile
    #pragma unroll
    for (int jj = sj; jj < 64; jj += 4)
      whT[sd * 66 + jj] = whBase[(size_t)(jb + jj) * F_];
    __syncthreads();
    #pragma unroll
    for (int j0 = 0; j0 < 64; j0 += 4) {
      const int jo = j0 + (krow << 1);
      v2f er2 = *(const v2f*)&ler[jb + jo];       // e_r[j], e_r[j+1] (aligned)
      const float xA0 = elA + er2.x, xA1 = elA + er2.y;
      const float xB0 = elB + er2.x, xB1 = elB + er2.y;
      v2f fA = { __expf(fmaxf(xA0, ALPHA_ * xA0) - miA),
                 __expf(fmaxf(xA1, ALPHA_ * xA1) - miA) };
      v2f fB = { __expf(fmaxf(xB0, ALPHA_ * xB0) - miB),
                 __expf(fmaxf(xB1, ALPHA_ * xB1) - miB) };
      v2f bf0 = *(const v2f*)&whT[lidx * 66 + jo];         // Wh[j..j+1][0..15]
      v2f bf1 = *(const v2f*)&whT[(16 + lidx) * 66 + jo];  // Wh[j..j+1][16..31]
      aA0 = __builtin_amdgcn_wmma_f32_16x16x4_f32(false, fA, false, bf0,
                                                  (short)0, aA0, false, false);
      aA1 = __builtin_amdgcn_wmma_f32_16x16x4_f32(false, fA, false, bf1,
                                                  (short)0, aA1, false, false);
      aB0 = __builtin_amdgcn_wmma_f32_16x16x4_f32(false, fB, false, bf0,
                                                  (short)0, aB0, false, false);
      aB1 = __builtin_amdgcn_wmma_f32_16x16x4_f32(false, fB, false, bf1,
                                                  (short)0, aB1, false, false);
    }
  }
  const int rowA = i0 + wave * 32 + (krow << 3);
  const int rowB = rowA + 16;
  #pragma unroll
  for (int v = 0; v < 8; ++v) {
    const float siA = sinv[(size_t)bh * N_ + rowA + v];    // 1/sum per row
    const float siB = sinv[(size_t)bh * N_ + rowB + v];
    hp[(size_t)(b * N_ + rowA + v) * F_ + hh * DH_ + lidx]      = aA0[v] * siA;
    hp[(size_t)(b * N_ + rowA + v) * F_ + hh * DH_ + 16 + lidx] = aA1[v] * siA;
    hp[(size_t)(b * N_ + rowB + v) * F_ + hh * DH_ + lidx]      = aB0[v] * siB;
    hp[(size_t)(b * N_ + rowB + v) * F_ + hh * DH_ + 16 + lidx] = aB1[v] * siB;
  }
}

// ---------------- K5: ELU + residual + LayerNorm ----------------------------
__global__ void __launch_bounds__(256)
k5_epilogue(const float* __restrict__ hp, const float* __restrict__ h,
            const float* __restrict__ gamma, const float* __restrict__ beta,
            float* __restrict__ out) {
  const int lane = threadIdx.x & 31;
  const int row = blockIdx.x * 8 + (threadIdx.x >> 5);
  const float* hpr = hp + (size_t)row * F_;
  const float* hr  = h  + (size_t)row * F_;
  float v[8];
  float s = 0.f;
  #pragma unroll
  for (int q = 0; q < 8; ++q) {
    int idx = q * 32 + lane;
    float x = hpr[idx];
    x = x > 0.f ? x : (__expf(x) - 1.0f);       // ELU(alpha=1)
    x += hr[idx];
    v[q] = x;
    s += x;
  }
  float mu = lane_sum(s) * (1.0f / F_);
  float var = 0.f;
  #pragma unroll
  for (int q = 0; q < 8; ++q) { float d = v[q] - mu; var += d * d; }
  var = lane_sum(var) * (1.0f / F_);
  float rs = rsqrtf(var + LN_EPS_);
  #pragma unroll
  for (int q = 0; q < 8; ++q) {
    int idx = q * 32 + lane;
    out[(size_t)row * F_ + idx] = (v[q] - mu) * rs * gamma[idx] + beta[idx];
  }
}

extern "C" void kernel_launch(void* const* d_in, const int* in_sizes, int n_in,
                              void* d_out, int out_size, void* d_ws, size_t ws_size,
                              hipStream_t stream) {
  const float*   h     = (const float*)d_in[0];
  const uint8_t* mask  = (const uint8_t*)d_in[1];  // jnp.bool_ -> 1 byte/elem
  const float*   W     = (const float*)d_in[2];
  const float*   a     = (const float*)d_in[3];
  const float*   gamma = (const float*)d_in[4];
  const float*   beta  = (const float*)d_in[5];
  float* out = (float*)d_out;

  float* ws   = (float*)d_ws;
  float* Wt   = ws;                    // 256*256
  float* Wh   = Wt + 65536;            // 8192*256
  float* e_l  = Wh + 2097152;          // 65536
  float* e_r  = e_l + 65536;           // mask-biased
  float* marr = e_r + 65536;
  float* sinv = marr + 65536;
  float* hp   = sinv + 65536;          // 8192*256

  k0_transpose<<<256, 256, 0, stream>>>(W, Wt);
  k1_gemm_wh<<<dim3(64, 8), 256, 0, stream>>>(h, Wt, Wh);
  k2_el_er<<<(B_ * N_ * H_) / 8, 256, 0, stream>>>(Wh, a, mask, e_l, e_r);
  k3_stats<<<(B_ * H_ * N_) / 8, 256, 0, stream>>>(e_l, e_r, marr, sinv);
  k4_attn_pv<<<B_ * H_ * (N_ / 128), 128, 0, stream>>>(Wh, e_l, e_r,
                                                       marr, sinv, hp);
  k5_epilogue<<<(B_ * N_) / 8, 256, 0, stream>>>(hp, h, gamma, beta, out);
}